// GCNEncoder_81003083203566
// MI455X (gfx1250) — compile-verified
//
#include <hip/hip_runtime.h>
#include <hip/hip_bf16.h>

// GCN 2-layer forward for MI455X (gfx1250, wave32).
// GEMMs use native fp32 WMMA (v_wmma_f32_16x16x4_f32) -> exact reference precision.
// Edge aggregation is L2-resident (192MB L2 >> 130MB working set) scatter-add via
// hardware global_atomic_add_f32.

typedef float v2f __attribute__((ext_vector_type(2)));
typedef float v8f __attribute__((ext_vector_type(8)));

#define IN_C 128   // input channels (== K for both GEMMs since 2*hid == 128)
#define HID2 128   // 2*hid (layer-1 output width)
#define HID  64    // hid   (layer-2 output width)

__device__ __forceinline__ void atomic_add_f32(float* p, float v) {
    // relaxed, device scope -> lowers to global_atomic_add_u32/f32 (no CAS loop)
    __hip_atomic_fetch_add(p, v, __ATOMIC_RELAXED, __HIP_MEMORY_SCOPE_AGENT);
}

// ---------------- utility fills ----------------
__global__ __launch_bounds__(256)
void fill_kernel(float* __restrict__ p, float v, long long n) {
    long long i = (long long)blockIdx.x * 256 + threadIdx.x;
    if (i < n) p[i] = v;
}

// ---------------- degree / norm ----------------
__global__ __launch_bounds__(256)
void degree_kernel(const int* __restrict__ dst, float* __restrict__ deg, long long E) {
    long long e = (long long)blockIdx.x * 256 + threadIdx.x;
    if (e < E) atomic_add_f32(&deg[dst[e]], 1.0f);
}

__global__ __launch_bounds__(256)
void rsqrt_kernel(const float* __restrict__ deg, float* __restrict__ dinv, int n) {
    int i = blockIdx.x * 256 + threadIdx.x;
    if (i < n) dinv[i] = rsqrtf(deg[i]);   // deg >= 1 due to self-loops
}

// ---------------- fp32 WMMA GEMM: C[M,NOUT] = A[M,128] @ B[128,NOUT] ----------------
// One wave per 16x16 output tile. K consumed 4 at a time by v_wmma_f32_16x16x4_f32.
// A-matrix VGPR layout (32-bit A 16x4): lane m = lane&15, lane-half kh = lane>>4
//   a[0] = A[m, k0 + 2*kh], a[1] = A[m, k0 + 2*kh + 1]
// B-matrix (4x16, row striped across lanes): b[v] = B[k0 + 2*kh + v, n], n = lane&15
// C/D (16x16 f32, 8 VGPRs): c[r] -> row (kh*8 + r), col n.
template <int NOUT>
__global__ __launch_bounds__(128)
void gemm_wmma_f32(const float* __restrict__ A, const float* __restrict__ B,
                   float* __restrict__ C, int M) {
    constexpr int K = 128;
    constexpr int NTILES = NOUT / 16;
    const int lane = threadIdx.x & 31;
    const int wglobal = blockIdx.x * 4 + (threadIdx.x >> 5);
    const int tm = wglobal / NTILES;
    const int tn = wglobal % NTILES;
    if (tm * 16 >= M) return;                 // wave-uniform: EXEC stays all-ones

    const int m  = lane & 15;
    const int kh = lane >> 4;                 // 0 or 1
    const int n  = lane & 15;
    const int col = tn * 16 + n;
    const float* arow = A + (size_t)(tm * 16 + m) * K;

    v8f acc = {};
#pragma unroll 8
    for (int k0 = 0; k0 < K; k0 += 4) {
        const int ka = k0 + kh * 2;
        const float2 av = *reinterpret_cast<const float2*>(arow + ka);  // 8B aligned
        v2f a; a[0] = av.x; a[1] = av.y;
        v2f b;
        b[0] = B[(size_t)ka * NOUT + col];
        b[1] = B[(size_t)(ka + 1) * NOUT + col];
        // 8 args: (neg_a, A, neg_b, B, c_mod, C, reuse_a, reuse_b)
        acc = __builtin_amdgcn_wmma_f32_16x16x4_f32(false, a, false, b,
                                                    (short)0, acc, false, false);
    }

    const int row0 = tm * 16 + kh * 8;
#pragma unroll
    for (int r = 0; r < 8; ++r)
        C[(size_t)(row0 + r) * NOUT + col] = acc[r];
}

// ---------------- edge aggregation: acc[dst] += h[src] * dinv[src]*dinv[dst] ----------------
// One thread per (edge, 4-feature chunk). A wave's 32 lanes cover one 128-float row
// (layer 1) => fully coalesced 512B gather + contiguous atomic burst into L2.
template <int F, int LOGC>   // F = feature width, LOGC = log2(F/4)
__global__ __launch_bounds__(256)
void edge_aggregate(const int* __restrict__ src, const int* __restrict__ dst,
                    const float* __restrict__ h, const float* __restrict__ dinv,
                    float* __restrict__ acc, long long total) {
    long long g = (long long)blockIdx.x * 256 + threadIdx.x;
    if (g >= total) return;
    const long long e = g >> LOGC;
    const int c = (int)(g & ((1 << LOGC) - 1));
    const int s = src[e];
    const int d = dst[e];
    const float norm = dinv[s] * dinv[d];
    const float4 hv = *reinterpret_cast<const float4*>(h + (size_t)s * F + c * 4);
    float* ap = acc + (size_t)d * F + c * 4;
    atomic_add_f32(ap + 0, hv.x * norm);
    atomic_add_f32(ap + 1, hv.y * norm);
    atomic_add_f32(ap + 2, hv.z * norm);
    atomic_add_f32(ap + 3, hv.w * norm);
}

// ---------------- finalize: out = [relu](acc + h*dinv^2 + bias) ----------------
// Self-loop contribution (dinv[i]^2 * h[i]) folded in here -> no atomics for loops.
template <int F, int LOGF, bool RELU>
__global__ __launch_bounds__(256)
void finalize_kernel(const float* __restrict__ acc, const float* __restrict__ h,
                     const float* __restrict__ dinv, const float* __restrict__ bias,
                     float* out, long long total) {
    long long i = (long long)blockIdx.x * 256 + threadIdx.x;
    if (i >= total) return;
    const int node = (int)(i >> LOGF);
    const int f = (int)(i & (F - 1));
    const float di = dinv[node];
    float v = acc[i] + h[i] * di * di + bias[f];
    out[i] = RELU ? fmaxf(v, 0.0f) : v;
}

static inline int blocks_for(long long n, int bs) { return (int)((n + bs - 1) / bs); }

extern "C" void kernel_launch(void* const* d_in, const int* in_sizes, int n_in,
                              void* d_out, int out_size, void* d_ws, size_t ws_size,
                              hipStream_t stream) {
    const float* x  = (const float*)d_in[0];
    const int*   ei = (const int*)d_in[1];
    const float* W1 = (const float*)d_in[2];
    const float* b1 = (const float*)d_in[3];
    const float* W2 = (const float*)d_in[4];
    const float* b2 = (const float*)d_in[5];
    float* out = (float*)d_out;

    const int N = in_sizes[0] / IN_C;               // 100000
    const long long E = (long long)in_sizes[1] / 2; // 1.6M
    const int* esrc = ei;
    const int* edst = ei + E;

    // workspace layout (floats): deg | dinv | h1[N*128] | acc1[N*128] | acc2[N*64]
    float* ws   = (float*)d_ws;
    float* deg  = ws;
    float* dinv = deg + N;
    float* h1   = dinv + N;
    float* acc1 = h1 + (size_t)N * HID2;
    float* acc2 = acc1 + (size_t)N * HID2;
    float* h2   = out;                               // stage layer-2 GEMM in d_out

    const long long n1 = (long long)N * HID2;        // 12.8M
    const long long n2 = (long long)N * HID;         // 6.4M

    // ---- init ----
    fill_kernel<<<blocks_for(n1, 256), 256, 0, stream>>>(acc1, 0.0f, n1);
    fill_kernel<<<blocks_for(n2, 256), 256, 0, stream>>>(acc2, 0.0f, n2);
    fill_kernel<<<blocks_for(N, 256), 256, 0, stream>>>(deg, 1.0f, N);   // self-loop

    // ---- degree + symmetric norm (shared by both layers) ----
    degree_kernel<<<blocks_for(E, 256), 256, 0, stream>>>(edst, deg, E);
    rsqrt_kernel<<<blocks_for(N, 256), 256, 0, stream>>>(deg, dinv, N);

    // ---- layer 1: h1 = x @ W1 ----
    {
        const int mtiles = (N + 15) / 16;
        const int waves = mtiles * (HID2 / 16);
        gemm_wmma_f32<HID2><<<(waves + 3) / 4, 128, 0, stream>>>(x, W1, h1, N);
    }
    edge_aggregate<HID2, 5><<<blocks_for(E * 32, 256), 256, 0, stream>>>(
        esrc, edst, h1, dinv, acc1, E * 32);
    finalize_kernel<HID2, 7, true><<<blocks_for(n1, 256), 256, 0, stream>>>(
        acc1, h1, dinv, b1, acc1, n1);               // in-place: acc1 = relu(gcn1)

    // ---- layer 2: h2 = relu_out @ W2 ----
    {
        const int mtiles = (N + 15) / 16;
        const int waves = mtiles * (HID / 16);
        gemm_wmma_f32<HID><<<(waves + 3) / 4, 128, 0, stream>>>(acc1, W2, h2, N);
    }
    edge_aggregate<HID, 4><<<blocks_for(E * 16, 256), 256, 0, stream>>>(
        esrc, edst, h2, dinv, acc2, E * 16);
    finalize_kernel<HID, 6, false><<<blocks_for(n2, 256), 256, 0, stream>>>(
        acc2, h2, dinv, b2, out, n2);                // h2 lives in d_out; final overwrite
}